// MetricSelfAttention_25838523253000
// MI455X (gfx1250) — compile-verified
//
#include <hip/hip_runtime.h>
#include <hip/hip_bf16.h>

// ---------------------------------------------------------------------------
// MetricSelfAttention on MI455X (gfx1250, wave32, WMMA).
//
// Algebra: the reference's softmax is dead code, so
//   nudged = (P M P^T) P == P (M (P^T P))
// removing the 537MB WxW score matrix (~34 of ~52 GFLOP). Remaining ~19 GFLOP
// runs on V_WMMA_F32_16X16X4_F32 (full fp32 = reference numerics).
//
// Round-2 changes vs round-1:
//  * All B operands staged TRANSPOSED in LDS (Bt[n][k], stride 68) so every
//    fragment load (A and B) is one contiguous ds_load_b64 per lane —
//    removes the ds_load_2addr_b32 + v_mov marshalling seen in round-1 asm.
//  * 2x2 register blocking (128x64 block tile, 4 accumulators/wave):
//    2 A-frags + 2 B-frags feed 4 WMMAs -> 1.0 DS loads per WMMA (was 1.5).
// ---------------------------------------------------------------------------

typedef __attribute__((ext_vector_type(2))) float v2f;
typedef __attribute__((ext_vector_type(8))) float v8f;

#define LDS_STRIDE 68   // 64 + 4 pad: fragment lanes step 4 banks -> conflict-free

static __device__ __forceinline__ v8f wmma_f32(v2f a, v2f b, v8f c) {
    return __builtin_amdgcn_wmma_f32_16x16x4_f32(
        /*neg_a=*/false, a, /*neg_b=*/false, b,
        /*c_mod=*/(short)0, c, /*reuse_a=*/false, /*reuse_b=*/false);
}

// Unified fragment load (A from row-major [m][k] tile, B from transposed
// [n][k] tile, both stride 68). Lane l reads 2 consecutive floats at
// row (l&15), K = kk + 2*(l>>4)  ->  single ds_load_b64, conflict-free.
static __device__ __forceinline__ v2f load_frag(const float* S, int lane) {
    return *reinterpret_cast<const v2f*>(S + (lane & 15) * LDS_STRIDE + 2 * (lane >> 4));
}

// C/D tile store: VGPR r of lane l = D[r + 8*(l>=16)][l&15]  (ISA 7.12.2)
static __device__ __forceinline__ void store_tile_d(float* g, int ldg, const v8f& d, int lane) {
    int n = lane & 15;
    int mbase = (lane >> 4) * 8;
#pragma unroll
    for (int r = 0; r < 8; ++r) g[(mbase + r) * (size_t)ldg + n] = d[r];
}

// 64x64 global -> LDS, row-major, float4, fully coalesced.
static __device__ __forceinline__ void load_tile64(float* lds, const float* g, int ldg, int t) {
    int r = t >> 2;
    int c = (t & 3) * 16;
    const float* src = g + (size_t)r * ldg + c;
    float* dst = lds + r * LDS_STRIDE + c;
#pragma unroll
    for (int i = 0; i < 4; ++i) {
        float4 v = *reinterpret_cast<const float4*>(src + 4 * i);
        *reinterpret_cast<float4*>(dst + 4 * i) = v;
    }
}

// 128x64 global -> LDS, row-major.
static __device__ __forceinline__ void load_tile128(float* lds, const float* g, int ldg, int t) {
    int r = t >> 1;
    int c = (t & 1) * 32;
    const float* src = g + (size_t)r * ldg + c;
    float* dst = lds + r * LDS_STRIDE + c;
#pragma unroll
    for (int i = 0; i < 8; ++i) {
        float4 v = *reinterpret_cast<const float4*>(src + 4 * i);
        *reinterpret_cast<float4*>(dst + 4 * i) = v;
    }
}

// 64x64 global -> LDS TRANSPOSED: lds[(c)*68 + r] = g[r][c].
// r = lane id within quarter -> per-slot store banks (4*(c+i)+t) all distinct.
static __device__ __forceinline__ void load_tile64_T(float* lds, const float* g, int ldg, int t) {
    int r = t & 63;            // source row
    int c = (t >> 6) * 16;     // source col base
    const float* src = g + (size_t)r * ldg + c;
#pragma unroll
    for (int i = 0; i < 4; ++i) {
        float4 v = *reinterpret_cast<const float4*>(src + 4 * i);
        lds[(c + 4 * i + 0) * LDS_STRIDE + r] = v.x;
        lds[(c + 4 * i + 1) * LDS_STRIDE + r] = v.y;
        lds[(c + 4 * i + 2) * LDS_STRIDE + r] = v.z;
        lds[(c + 4 * i + 3) * LDS_STRIDE + r] = v.w;
    }
}

// ---------------------------------------------------------------------------
// Kernel 1: proj[bn][w][k] = X[b] @ Pn    (32 GEMMs [2048x1024]x[1024x64])
// 128x64 block tile; grid = 16 w-tiles * 32 bn = 512 blocks.
// ---------------------------------------------------------------------------
__global__ __launch_bounds__(256) void msa_proj_kernel(
    const float* __restrict__ X, const float* __restrict__ P, float* __restrict__ proj_ws) {
    __shared__ float Xs[128 * LDS_STRIDE];
    __shared__ float Pt[64 * LDS_STRIDE];     // transposed: Pt[k][c]
    const int t = threadIdx.x, lane = t & 31, wv = t >> 5;
    const int wt = blockIdx.x & 15;           // 2048/128
    const int bn = blockIdx.x >> 4;
    const int b = bn >> 4, n = bn & 15;
    const float* Xg = X + ((size_t)b * 2048 + (size_t)wt * 128) * 1024;
    const float* Pg = P + (size_t)n * 1024 * 64;
    const int mi0 = (wv & 3) * 2;             // M tiles mi0, mi0+1 (of 8)
    const int n0 = (wv >> 2) * 32;            // N cols n0, n0+16
    v8f a00 = {}, a01 = {}, a10 = {}, a11 = {};
    for (int cc = 0; cc < 1024; cc += 64) {
        __syncthreads();
        load_tile128(Xs, Xg + cc, 1024, t);
        load_tile64_T(Pt, Pg + (size_t)cc * 64, 64, t);
        __syncthreads();
#pragma unroll
        for (int kk = 0; kk < 64; kk += 4) {
            v2f fa0 = load_frag(Xs + mi0 * 16 * LDS_STRIDE + kk, lane);
            v2f fa1 = load_frag(Xs + (mi0 + 1) * 16 * LDS_STRIDE + kk, lane);
            v2f fb0 = load_frag(Pt + n0 * LDS_STRIDE + kk, lane);
            v2f fb1 = load_frag(Pt + (n0 + 16) * LDS_STRIDE + kk, lane);
            a00 = wmma_f32(fa0, fb0, a00);
            a01 = wmma_f32(fa0, fb1, a01);
            a10 = wmma_f32(fa1, fb0, a10);
            a11 = wmma_f32(fa1, fb1, a11);
        }
    }
    float* Og = proj_ws + ((size_t)bn * 2048 + wt * 128 + mi0 * 16) * 64;
    store_tile_d(Og + n0,            64, a00, lane);
    store_tile_d(Og + n0 + 16,       64, a01, lane);
    store_tile_d(Og + 16 * 64 + n0,       64, a10, lane);
    store_tile_d(Og + 16 * 64 + n0 + 16,  64, a11, lane);
}

// ---------------------------------------------------------------------------
// Kernel 2: G[bn] = P^T P   ([64x2048]x[2048x64]).
// One transposed tile Pt[j][w] serves BOTH operands (A[m][w]=P[w][m],
// Bt[n][w]=P[w][n]).  grid = 32 blocks.
// ---------------------------------------------------------------------------
__global__ __launch_bounds__(256) void msa_gram_kernel(
    const float* __restrict__ proj_ws, float* __restrict__ G_ws) {
    __shared__ float Pt[64 * LDS_STRIDE];
    const int t = threadIdx.x, lane = t & 31, wv = t >> 5;
    const int bn = blockIdx.x;
    const float* Pg = proj_ws + (size_t)bn * 2048 * 64;
    const int mi = wv & 3;                    // M tile (of 4)
    const int n0 = (wv >> 2) * 32;            // N cols n0, n0+16
    v8f acc0 = {}, acc1 = {};
    for (int ww = 0; ww < 2048; ww += 64) {
        __syncthreads();
        load_tile64_T(Pt, Pg + (size_t)ww * 64, 64, t);
        __syncthreads();
#pragma unroll
        for (int kk = 0; kk < 64; kk += 4) {
            v2f fa  = load_frag(Pt + mi * 16 * LDS_STRIDE + kk, lane);
            v2f fb0 = load_frag(Pt + n0 * LDS_STRIDE + kk, lane);
            v2f fb1 = load_frag(Pt + (n0 + 16) * LDS_STRIDE + kk, lane);
            acc0 = wmma_f32(fa, fb0, acc0);
            acc1 = wmma_f32(fa, fb1, acc1);
        }
    }
    float* Og = G_ws + (size_t)bn * 4096 + mi * 16 * 64;
    store_tile_d(Og + n0,      64, acc0, lane);
    store_tile_d(Og + n0 + 16, 64, acc1, lane);
}

// ---------------------------------------------------------------------------
// Kernel 3: A[bn] = M_n @ G[bn]  (17 MFLOP total, plain VALU).
// M on the fly: diag + symmetric triu(halves); idx(i<j)=i*(127-i)/2+(j-i-1).
// ---------------------------------------------------------------------------
__global__ __launch_bounds__(256) void msa_mg_kernel(
    const float* __restrict__ halves, const float* __restrict__ diag,
    const float* __restrict__ G_ws, float* __restrict__ A_ws) {
    const int bn = blockIdx.x, n = bn & 15;
    const int t = threadIdx.x;
    const int i  = t >> 2;
    const int j0 = (t & 3) * 16;
    const float* G  = G_ws + (size_t)bn * 4096;
    const float* hv = halves + (size_t)n * (64 * 63 / 2);
    const float* dg = diag + (size_t)n * 64;
    float acc[16];
#pragma unroll
    for (int j = 0; j < 16; ++j) acc[j] = 0.f;
    for (int k = 0; k < 64; ++k) {
        float m;
        if (k == i) {
            m = dg[i];
        } else {
            int a = i < k ? i : k;
            int c = i < k ? k : i;
            m = hv[a * (127 - a) / 2 + (c - a - 1)];
        }
        const float* Gr = G + k * 64 + j0;
#pragma unroll
        for (int j = 0; j < 16; ++j) acc[j] += m * Gr[j];
    }
    float* O = A_ws + (size_t)bn * 4096 + i * 64 + j0;
#pragma unroll
    for (int j = 0; j < 16; ++j) O[j] = acc[j];
}

// ---------------------------------------------------------------------------
// Kernel 4: nudged[b][w][n*64+k] = P[bn] @ A[bn]   ([2048x64]x[64x64])
// 128x64 block tile, single K chunk; grid = 16 w-tiles * 32 bn = 512 blocks.
// ---------------------------------------------------------------------------
__global__ __launch_bounds__(256) void msa_nudge_kernel(
    const float* __restrict__ proj_ws, const float* __restrict__ A_ws,
    float* __restrict__ nudged) {
    __shared__ float Ps[128 * LDS_STRIDE];
    __shared__ float At[64 * LDS_STRIDE];     // transposed: At[kout][kin]
    const int t = threadIdx.x, lane = t & 31, wv = t >> 5;
    const int wt = blockIdx.x & 15;
    const int bn = blockIdx.x >> 4;
    const int b = bn >> 4, n = bn & 15;
    const float* Pg = proj_ws + ((size_t)bn * 2048 + wt * 128) * 64;
    const float* Ag = A_ws + (size_t)bn * 4096;
    load_tile128(Ps, Pg, 64, t);
    load_tile64_T(At, Ag, 64, t);
    __syncthreads();
    const int mi0 = (wv & 3) * 2;
    const int n0 = (wv >> 2) * 32;
    v8f a00 = {}, a01 = {}, a10 = {}, a11 = {};
#pragma unroll
    for (int kk = 0; kk < 64; kk += 4) {
        v2f fa0 = load_frag(Ps + mi0 * 16 * LDS_STRIDE + kk, lane);
        v2f fa1 = load_frag(Ps + (mi0 + 1) * 16 * LDS_STRIDE + kk, lane);
        v2f fb0 = load_frag(At + n0 * LDS_STRIDE + kk, lane);
        v2f fb1 = load_frag(At + (n0 + 16) * LDS_STRIDE + kk, lane);
        a00 = wmma_f32(fa0, fb0, a00);
        a01 = wmma_f32(fa0, fb1, a01);
        a10 = wmma_f32(fa1, fb0, a10);
        a11 = wmma_f32(fa1, fb1, a11);
    }
    float* Og = nudged + ((size_t)b * 2048 + wt * 128 + mi0 * 16) * 1024 + n * 64;
    store_tile_d(Og + n0,                 1024, a00, lane);
    store_tile_d(Og + n0 + 16,            1024, a01, lane);
    store_tile_d(Og + 16 * 1024 + n0,      1024, a10, lane);
    store_tile_d(Og + 16 * 1024 + n0 + 16, 1024, a11, lane);
}

// ---------------------------------------------------------------------------
// Kernel 5: out = nudged @ W^T + bias   ([4096x1024]x[1024x1024]^T)
// Bt[n][k] = Wm[j0+n][c0+k]: the row-major Wm tile IS the transposed-B
// layout, so no transpose needed.  grid = 32 row-tiles * 16 col-tiles = 512.
// ---------------------------------------------------------------------------
__global__ __launch_bounds__(256) void msa_mixer_kernel(
    const float* __restrict__ Ng, const float* __restrict__ Wm,
    const float* __restrict__ bias, float* __restrict__ out) {
    __shared__ float Ns[128 * LDS_STRIDE];
    __shared__ float Ws[64 * LDS_STRIDE];     // Ws[j][c]
    const int t = threadIdx.x, lane = t & 31, wv = t >> 5;
    const int jt = blockIdx.x & 15;           // 1024/64 col tiles
    const int rt = blockIdx.x >> 4;           // 4096/128 row tiles
    const float* Nbase = Ng + (size_t)rt * 128 * 1024;
    const float* Wbase = Wm + (size_t)jt * 64 * 1024;
    const int mi0 = (wv & 3) * 2;
    const int n0 = (wv >> 2) * 32;
    v8f a00 = {}, a01 = {}, a10 = {}, a11 = {};
    for (int cc = 0; cc < 1024; cc += 64) {
        __syncthreads();
        load_tile128(Ns, Nbase + cc, 1024, t);
        load_tile64(Ws, Wbase + cc, 1024, t);
        __syncthreads();
#pragma unroll
        for (int kk = 0; kk < 64; kk += 4) {
            v2f fa0 = load_frag(Ns + mi0 * 16 * LDS_STRIDE + kk, lane);
            v2f fa1 = load_frag(Ns + (mi0 + 1) * 16 * LDS_STRIDE + kk, lane);
            v2f fb0 = load_frag(Ws + n0 * LDS_STRIDE + kk, lane);
            v2f fb1 = load_frag(Ws + (n0 + 16) * LDS_STRIDE + kk, lane);
            a00 = wmma_f32(fa0, fb0, a00);
            a01 = wmma_f32(fa0, fb1, a01);
            a10 = wmma_f32(fa1, fb0, a10);
            a11 = wmma_f32(fa1, fb1, a11);
        }
    }
    const float bv0 = bias[jt * 64 + n0 + (lane & 15)];
    const float bv1 = bias[jt * 64 + n0 + 16 + (lane & 15)];
#pragma unroll
    for (int r = 0; r < 8; ++r) { a00[r] += bv0; a01[r] += bv1; a10[r] += bv0; a11[r] += bv1; }
    float* Og = out + ((size_t)rt * 128 + mi0 * 16) * 1024 + jt * 64;
    store_tile_d(Og + n0,                 1024, a00, lane);
    store_tile_d(Og + n0 + 16,            1024, a01, lane);
    store_tile_d(Og + 16 * 1024 + n0,      1024, a10, lane);
    store_tile_d(Og + 16 * 1024 + n0 + 16, 1024, a11, lane);
}

// ---------------------------------------------------------------------------
extern "C" void kernel_launch(void* const* d_in, const int* in_sizes, int n_in,
                              void* d_out, int out_size, void* d_ws, size_t ws_size,
                              hipStream_t stream) {
    (void)in_sizes; (void)n_in; (void)out_size; (void)ws_size;
    const float* X      = (const float*)d_in[0];   // [2,2048,1024]
    const float* P      = (const float*)d_in[1];   // [1,16,1024,64]
    const float* halves = (const float*)d_in[2];   // [16, 2016]
    const float* diag   = (const float*)d_in[3];   // [16, 64]
    const float* Wm     = (const float*)d_in[4];   // [1024,1024]
    const float* bias   = (const float*)d_in[5];   // [1024]
    float* out = (float*)d_out;                    // [2,2048,1024]

    float* ws      = (float*)d_ws;
    float* proj_ws = ws;                       // 32*2048*64      = 4,194,304 f
    float* G_ws    = proj_ws + 4194304;        // 32*64*64        =   131,072 f
    float* A_ws    = G_ws + 131072;            // 32*64*64        =   131,072 f
    float* nudged  = A_ws + 131072;            // 2*2048*1024     = 4,194,304 f

    msa_proj_kernel <<<dim3(512), dim3(256), 0, stream>>>(X, P, proj_ws);
    msa_gram_kernel <<<dim3(32),  dim3(256), 0, stream>>>(proj_ws, G_ws);
    msa_mg_kernel   <<<dim3(32),  dim3(256), 0, stream>>>(halves, diag, G_ws, A_ws);
    msa_nudge_kernel<<<dim3(512), dim3(256), 0, stream>>>(proj_ws, A_ws, nudged);
    msa_mixer_kernel<<<dim3(512), dim3(256), 0, stream>>>(nudged, Wm, bias, out);
}